// WootCharacterWithQuat_53429393162751
// MI455X (gfx1250) — compile-verified
//
#include <hip/hip_runtime.h>

// ---------------------------------------------------------------------------
// Problem constants
// ---------------------------------------------------------------------------
static constexpr int Bb = 8;        // batches
static constexpr int Vv = 262144;   // vertices
static constexpr int Jj = 64;       // joints
static constexpr int Kk = 4;        // skin links per vertex
static constexpr int Gg = 2048;     // graph nodes
static constexpr int Ll = 8;        // graph links per vertex
static constexpr int Ee = 1835008;  // laplacian edges
static constexpr float EPS = 1e-8f;

// ---------------------------------------------------------------------------
// gfx1250 async global->LDS copy.
// Probe round revealed: param0 is global (AS1) 'v4i*' ("int __device__ *"
// with vector_size(16) in the diagnostic). Param1 is the LDS (AS3) dest.
// ---------------------------------------------------------------------------
#if __has_builtin(__builtin_amdgcn_global_load_async_to_lds_b128)
#define HAVE_ASYNC_LDS 1
typedef int v4i __attribute__((ext_vector_type(4)));
typedef __attribute__((address_space(1))) v4i gbl_v4i;
typedef __attribute__((address_space(3))) v4i lds_v4i;
__device__ __forceinline__ void async_cp16(const float* g, float* s) {
  __builtin_amdgcn_global_load_async_to_lds_b128((gbl_v4i*)g, (lds_v4i*)s,
                                                 /*offset=*/0, /*cpol=*/0);
}
__device__ __forceinline__ void async_wait0() {
#if __has_builtin(__builtin_amdgcn_s_wait_asynccnt)
  __builtin_amdgcn_s_wait_asynccnt(0);
#else
  asm volatile("s_wait_asynccnt 0" ::: "memory");
#endif
}
#else
#define HAVE_ASYNC_LDS 0
#endif

// Cooperative stage of nfloats (multiple of 4, 16B-aligned) into LDS.
__device__ __forceinline__ void stage(const float* __restrict__ g, float* s,
                                      int nfloats) {
  const int nvec = nfloats >> 2;
  for (int i = threadIdx.x; i < nvec; i += blockDim.x) {
#if HAVE_ASYNC_LDS
    async_cp16(g + 4 * i, s + 4 * i);
#else
    ((float4*)s)[i] = ((const float4*)g)[i];
#endif
  }
}

__device__ __forceinline__ void stage_done() {
#if HAVE_ASYNC_LDS
  async_wait0();
#endif
}

// ---------------------------------------------------------------------------
// Kernel 0: normalize joint quats (+ dual part) and node quats into workspace
//   ws layout (floats): jqn[B*J*4] | jd[B*J*4] | nqn[B*G*4]
// ---------------------------------------------------------------------------
__global__ void prep_kernel(const float* __restrict__ joint_quat,
                            const float* __restrict__ joint_trans,
                            const float* __restrict__ node_quat,
                            float* __restrict__ jqn, float* __restrict__ jd,
                            float* __restrict__ nqn) {
  int i = blockIdx.x * blockDim.x + threadIdx.x;
  if (i < Bb * Gg) {
    float qw = node_quat[4 * i + 0], qx = node_quat[4 * i + 1];
    float qy = node_quat[4 * i + 2], qz = node_quat[4 * i + 3];
    float inv = 1.0f / (sqrtf(qw * qw + qx * qx + qy * qy + qz * qz) + EPS);
    nqn[4 * i + 0] = qw * inv;
    nqn[4 * i + 1] = qx * inv;
    nqn[4 * i + 2] = qy * inv;
    nqn[4 * i + 3] = qz * inv;
  }
  if (i < Bb * Jj) {
    float qw = joint_quat[4 * i + 0], qx = joint_quat[4 * i + 1];
    float qy = joint_quat[4 * i + 2], qz = joint_quat[4 * i + 3];
    float inv = 1.0f / (sqrtf(qw * qw + qx * qx + qy * qy + qz * qz) + EPS);
    qw *= inv; qx *= inv; qy *= inv; qz *= inv;
    jqn[4 * i + 0] = qw;
    jqn[4 * i + 1] = qx;
    jqn[4 * i + 2] = qy;
    jqn[4 * i + 3] = qz;
    float tx = joint_trans[3 * i + 0], ty = joint_trans[3 * i + 1],
          tz = joint_trans[3 * i + 2];
    // jd = 0.5 * qmul((0, t), q):  w = -t.qv ; v = qw*t + cross(t, qv)
    jd[4 * i + 0] = -0.5f * (tx * qx + ty * qy + tz * qz);
    jd[4 * i + 1] = 0.5f * (qw * tx + (ty * qz - tz * qy));
    jd[4 * i + 2] = 0.5f * (qw * ty + (tz * qx - tx * qz));
    jd[4 * i + 3] = 0.5f * (qw * tz + (tx * qy - ty * qx));
  }
}

// ---------------------------------------------------------------------------
// Kernel 1: posed vertices. One thread per vertex, batch loop inside.
// Per-batch node/joint tables staged into LDS (CDNA5: 320 KB/WGP).
// ---------------------------------------------------------------------------
static constexpr int POSED_BLOCK = 1024;
// LDS layout (floats): gp[3G] | nq[4G] | nt[3G] | jq[4J] | jd[4J]
static constexpr int SM_GP = 0;
static constexpr int SM_NQ = SM_GP + 3 * Gg;
static constexpr int SM_NT = SM_NQ + 4 * Gg;
static constexpr int SM_JQ = SM_NT + 3 * Gg;
static constexpr int SM_JD = SM_JQ + 4 * Jj;
static constexpr int SM_TOTAL = SM_JD + 4 * Jj;  // 20992 floats = 83968 B

__global__ __launch_bounds__(POSED_BLOCK) void posed_kernel(
    const float* __restrict__ temp_verts, const float* __restrict__ skin_w,
    const float* __restrict__ node_pos, const float* __restrict__ node_trans,
    const float* __restrict__ link_w, const int* __restrict__ skin_idx,
    const int* __restrict__ link_idx, const float* __restrict__ jqn,
    const float* __restrict__ jdual, const float* __restrict__ nqn,
    float* __restrict__ posed_out) {
  extern __shared__ float smem[];
  float* s_gp = smem + SM_GP;
  float* s_nq = smem + SM_NQ;
  float* s_nt = smem + SM_NT;
  float* s_jq = smem + SM_JQ;
  float* s_jd = smem + SM_JD;

  const int v = blockIdx.x * POSED_BLOCK + threadIdx.x;

  // node positions are batch-independent: stage once
  stage(node_pos, s_gp, 3 * Gg);

  // per-vertex static data (read once, reused for all 8 batches)
  const float tvx = temp_verts[3 * v + 0];
  const float tvy = temp_verts[3 * v + 1];
  const float tvz = temp_verts[3 * v + 2];

  float lw[Ll];
  int li[Ll];
  float lsum = 0.f;
#pragma unroll
  for (int l = 0; l < Ll; ++l) {
    lw[l] = link_w[v * Ll + l];
    li[l] = link_idx[v * Ll + l];
    lsum += lw[l];
  }
  const float linv = 1.0f / (lsum + EPS);
#pragma unroll
  for (int l = 0; l < Ll; ++l) lw[l] *= linv;

  float sw[Kk];
  int si[Kk];
  float ssum = 0.f;
#pragma unroll
  for (int k = 0; k < Kk; ++k) {
    sw[k] = skin_w[v * Kk + k];
    si[k] = skin_idx[v * Kk + k];
    ssum += sw[k];
  }
  const float sinv = 1.0f / (ssum + EPS);
#pragma unroll
  for (int k = 0; k < Kk; ++k) sw[k] *= sinv;

  for (int b = 0; b < Bb; ++b) {
    __syncthreads();  // previous batch finished reading LDS
    stage(nqn + (size_t)b * Gg * 4, s_nq, 4 * Gg);
    stage(node_trans + (size_t)b * Gg * 3, s_nt, 3 * Gg);
    stage(jqn + b * Jj * 4, s_jq, 4 * Jj);
    stage(jdual + b * Jj * 4, s_jd, 4 * Jj);
    stage_done();
    __syncthreads();

    // ---- embedded-graph warp: v_eg = sum_l lw[l] * (qrot(nq, tv-gp)+gp+nt)
    float ex = 0.f, ey = 0.f, ez = 0.f;
#pragma unroll
    for (int l = 0; l < Ll; ++l) {
      const int g = li[l];
      const float gx = s_gp[3 * g + 0], gy = s_gp[3 * g + 1],
                  gz = s_gp[3 * g + 2];
      const float qw = s_nq[4 * g + 0], qx = s_nq[4 * g + 1],
                  qy = s_nq[4 * g + 2], qz = s_nq[4 * g + 3];
      const float lx = tvx - gx, ly = tvy - gy, lz = tvz - gz;
      // qrot: t = cross(qv, l) + qw*l ; r = l + 2*cross(qv, t)
      const float tx = (qy * lz - qz * ly) + qw * lx;
      const float ty = (qz * lx - qx * lz) + qw * ly;
      const float tz = (qx * ly - qy * lx) + qw * lz;
      const float rx = lx + 2.f * (qy * tz - qz * ty);
      const float ry = ly + 2.f * (qz * tx - qx * tz);
      const float rz = lz + 2.f * (qx * ty - qy * tx);
      ex = fmaf(lw[l], rx + gx + s_nt[3 * g + 0], ex);
      ey = fmaf(lw[l], ry + gy + s_nt[3 * g + 1], ey);
      ez = fmaf(lw[l], rz + gz + s_nt[3 * g + 2], ez);
    }

    // ---- dual-quaternion blend
    const int j0 = si[0];
    const float r0w = s_jq[4 * j0 + 0], r0x = s_jq[4 * j0 + 1],
                r0y = s_jq[4 * j0 + 2], r0z = s_jq[4 * j0 + 3];
    float rbw = 0.f, rbx = 0.f, rby = 0.f, rbz = 0.f;
    float dbw = 0.f, dbx = 0.f, dby = 0.f, dbz = 0.f;
#pragma unroll
    for (int k = 0; k < Kk; ++k) {
      const int j = si[k];
      const float rw = s_jq[4 * j + 0], rx = s_jq[4 * j + 1],
                  ry = s_jq[4 * j + 2], rz = s_jq[4 * j + 3];
      const float dw = s_jd[4 * j + 0], dx = s_jd[4 * j + 1],
                  dy = s_jd[4 * j + 2], dz = s_jd[4 * j + 3];
      const float dot = rw * r0w + rx * r0x + ry * r0y + rz * r0z;
      const float w = (dot >= 0.f) ? sw[k] : -sw[k];
      rbw = fmaf(w, rw, rbw); rbx = fmaf(w, rx, rbx);
      rby = fmaf(w, ry, rby); rbz = fmaf(w, rz, rbz);
      dbw = fmaf(w, dw, dbw); dbx = fmaf(w, dx, dbx);
      dby = fmaf(w, dy, dby); dbz = fmaf(w, dz, dbz);
    }
    const float inv =
        1.0f / (sqrtf(rbw * rbw + rbx * rbx + rby * rby + rbz * rbz) + EPS);
    rbw *= inv; rbx *= inv; rby *= inv; rbz *= inv;
    dbw *= inv; dbx *= inv; dby *= inv; dbz *= inv;

    // trans = 2*(rw*dv - dw*rv + cross(rv, dv))
    const float trx = 2.f * (rbw * dbx - dbw * rbx + (rby * dbz - rbz * dby));
    const float try_ = 2.f * (rbw * dby - dbw * rby + (rbz * dbx - rbx * dbz));
    const float trz = 2.f * (rbw * dbz - dbw * rbz + (rbx * dby - rby * dbx));

    // posed = qrot(rb, v_eg) + trans
    const float c1x = (rby * ez - rbz * ey) + rbw * ex;
    const float c1y = (rbz * ex - rbx * ez) + rbw * ey;
    const float c1z = (rbx * ey - rby * ex) + rbw * ez;
    const float px = ex + 2.f * (rby * c1z - rbz * c1y) + trx;
    const float py = ey + 2.f * (rbz * c1x - rbx * c1z) + try_;
    const float pz = ez + 2.f * (rbx * c1y - rby * c1x) + trz;

    float* o = posed_out + ((size_t)b * Vv + v) * 3;
    o[0] = px;
    o[1] = py;
    o[2] = pz;
  }
}

// ---------------------------------------------------------------------------
// Kernel 2: zero the laplacian output region (must happen every call)
// ---------------------------------------------------------------------------
__global__ void zero_kernel(float* __restrict__ p, int n) {
  int i = blockIdx.x * blockDim.x + threadIdx.x;
  if (i < n) p[i] = 0.f;
}

// ---------------------------------------------------------------------------
// Kernel 3: edge-list laplacian scatter. posed + lap are L2-resident (50 MB
// vs 192 MB L2), so gathers/atomics stay on-chip. Edge data read once,
// reused across all 8 batches. Prefetch next stride (global_prefetch_b8).
// ---------------------------------------------------------------------------
static constexpr int LAP_BLOCK = 256;
static constexpr int LAP_ITERS = 2;

__global__ __launch_bounds__(LAP_BLOCK) void lap_kernel(
    const float* __restrict__ posed, const float* __restrict__ lap_w,
    const int* __restrict__ lap_st, const int* __restrict__ lap_ed,
    float* __restrict__ lap_out) {
  const int stride = gridDim.x * LAP_BLOCK;
  int e = blockIdx.x * LAP_BLOCK + threadIdx.x;
  for (int it = 0; it < LAP_ITERS && e < Ee; ++it, e += stride) {
    if (e + stride < Ee) {
      __builtin_prefetch(lap_st + e + stride, 0, 1);
      __builtin_prefetch(lap_ed + e + stride, 0, 1);
      __builtin_prefetch(lap_w + e + stride, 0, 1);
    }
    const int s = lap_st[e];
    const int d = lap_ed[e];
    const float w = lap_w[e];
#pragma unroll
    for (int b = 0; b < Bb; ++b) {
      const float* p = posed + ((size_t)b * Vv + d) * 3;
      float* o = lap_out + ((size_t)b * Vv + s) * 3;
      atomicAdd(o + 0, w * p[0]);
      atomicAdd(o + 1, w * p[1]);
      atomicAdd(o + 2, w * p[2]);
    }
  }
}

// ---------------------------------------------------------------------------
// Host launcher
// ---------------------------------------------------------------------------
extern "C" void kernel_launch(void* const* d_in, const int* in_sizes, int n_in,
                              void* d_out, int out_size, void* d_ws,
                              size_t ws_size, hipStream_t stream) {
  const float* temp_verts  = (const float*)d_in[0];
  const float* joint_quat  = (const float*)d_in[1];
  const float* joint_trans = (const float*)d_in[2];
  const float* skin_w      = (const float*)d_in[3];
  const float* node_pos    = (const float*)d_in[4];
  const float* node_quat   = (const float*)d_in[5];
  const float* node_trans  = (const float*)d_in[6];
  const float* link_w      = (const float*)d_in[7];
  const float* lap_w       = (const float*)d_in[8];
  const int*   skin_idx    = (const int*)d_in[9];
  const int*   link_idx    = (const int*)d_in[10];
  const int*   lap_st      = (const int*)d_in[11];
  const int*   lap_ed      = (const int*)d_in[12];

  float* out = (float*)d_out;
  float* posed = out;                           // (B, V, 3)
  float* lap = out + (size_t)Bb * Vv * 3;       // (B, V, 3)

  float* ws  = (float*)d_ws;
  float* jqn = ws;                       // B*J*4
  float* jd  = jqn + Bb * Jj * 4;        // B*J*4
  float* nqn = jd + Bb * Jj * 4;         // B*G*4

  // 0) quat normalization + dual parts
  {
    int n = Bb * Gg;  // covers B*J too
    prep_kernel<<<dim3((n + 255) / 256), dim3(256), 0, stream>>>(
        joint_quat, joint_trans, node_quat, jqn, jd, nqn);
  }

  // 1) posed vertices (82 KB dynamic LDS per workgroup — CDNA5 only)
  {
    size_t smem_bytes = (size_t)SM_TOTAL * sizeof(float);
    posed_kernel<<<dim3(Vv / POSED_BLOCK), dim3(POSED_BLOCK), smem_bytes,
                   stream>>>(temp_verts, skin_w, node_pos, node_trans, link_w,
                             skin_idx, link_idx, jqn, jd, nqn, posed);
  }

  // 2) zero laplacian accumulator
  {
    int n = Bb * Vv * 3;
    zero_kernel<<<dim3((n + 255) / 256), dim3(256), 0, stream>>>(lap, n);
  }

  // 3) laplacian edge scatter
  {
    int blocks = (Ee + LAP_BLOCK * LAP_ITERS - 1) / (LAP_BLOCK * LAP_ITERS);
    lap_kernel<<<dim3(blocks), dim3(LAP_BLOCK), 0, stream>>>(posed, lap_w,
                                                             lap_st, lap_ed,
                                                             lap);
  }
}